// IdealDiffuser_80461917323320
// MI455X (gfx1250) — compile-verified
//
#include <hip/hip_runtime.h>
#include <math.h>

#define B_N   1024
#define M_N   50000
#define D_N   512
#define DC_N  16
#define MT    32          // m-tile rows
#define BT    16          // b-tile cols
#define SP    516         // LDS row stride (floats) for D=512 tiles (bank-friendly)
#define CP    20          // LDS row stride for DC=16 tiles
#define WJP   34          // weight-transpose row stride
#define NCH1  16          // M chunks for stats pass
#define NCH3  8           // M chunks for accumulate pass
#define NMT   ((M_N + MT - 1) / MT)          // 1563 m-tiles
#define CPT1  ((NMT + NCH1 - 1) / NCH1)      // tiles per chunk, pass 1
#define CPT3  ((NMT + NCH3 - 1) / NCH3)      // tiles per chunk, pass 3

typedef float v2f __attribute__((ext_vector_type(2)));
typedef float v8f __attribute__((ext_vector_type(8)));

__device__ __forceinline__ v8f wmma4(v2f a, v2f b, v8f c) {
  // D(16x16,f32) = A(16x4,f32) * B(4x16,f32) + C   -> v_wmma_f32_16x16x4_f32
  return __builtin_amdgcn_wmma_f32_16x16x4_f32(false, a, false, b, (short)0, c, false, false);
}

struct __align__(16) SmemT {
  float xs[BT * SP];      // x tile   [16][512] padded
  float smp[MT * SP];     // samples  [32][512] padded
  float condx[BT * CP];   // cond     [16][16]  padded
  float cd[MT * CP];      // cond_data[32][16]  padded
  float S[MT * BT];       // distance GEMM tile (reduced across waves)
  float Sc[MT * BT];      // cond GEMM tile
  float a2[MT];           // ||samples_m||^2
  float ca2[MT];          // ||cond_data_m||^2
  float inv2s2[BT];       // 1/(2 sigma^2)
  float Wt[BT * WJP];     // weights transposed [b][j]  (pass 3)
  float fmaxs[BT];
  float finvZ[BT];
};

// ---- shared helpers ----------------------------------------------------

__device__ __forceinline__ void load_tiles(SmemT& sm, const float* __restrict__ samples,
                                           const float* __restrict__ cond_data,
                                           int m0, int t) {
  // zero the cross-wave reduction buffers (consumed after next barrier)
  for (int i = t; i < MT * BT; i += 256) { sm.S[i] = 0.0f; sm.Sc[i] = 0.0f; }
  if (t < MT) { sm.a2[t] = 0.0f; sm.ca2[t] = 0.0f; }
  // samples tile: 32x512 f32 = 4096 float4, fully coalesced (rows contiguous)
#pragma unroll
  for (int i = 0; i < 16; ++i) {
    int fi  = i * 256 + t;
    int row = fi >> 7;          // 128 float4 per row
    int c4  = fi & 127;
    float4 v = make_float4(0.f, 0.f, 0.f, 0.f);
    if (m0 + row < M_N)
      v = *(const float4*)(samples + (size_t)(m0 + row) * D_N + c4 * 4);
    *(float4*)(&sm.smp[row * SP + c4 * 4]) = v;
  }
  // cond_data tile: 32x16 = 128 float4
  if (t < 128) {
    int row = t >> 2, c4 = t & 3;
    float4 v = make_float4(0.f, 0.f, 0.f, 0.f);
    if (m0 + row < M_N)
      v = *(const float4*)(cond_data + (size_t)(m0 + row) * DC_N + c4 * 4);
    *(float4*)(&sm.cd[row * CP + c4 * 4]) = v;
  }
}

__device__ __forceinline__ void compute_norms(SmemT& sm, int t) {
  { // ||samples_m||^2 : 8 threads per row, 64 floats each
    int row = t >> 3, seg = t & 7;
    const float* p = &sm.smp[row * SP + seg * 64];
    float s = 0.0f;
#pragma unroll
    for (int i = 0; i < 64; ++i) s += p[i] * p[i];
    unsafeAtomicAdd(&sm.a2[row], s);       // ds_add_f32
  }
  if (t < MT) { // ||cond_data_m||^2
    const float* p = &sm.cd[t * CP];
    float s = 0.0f;
#pragma unroll
    for (int i = 0; i < DC_N; ++i) s += p[i] * p[i];
    sm.ca2[t] = s;                          // single writer
  }
}

// Distance GEMM: S[m][b] += sum_k samples[m][k] * x[b][k], wave wv owns K-slice [64*wv, 64*wv+64)
__device__ __forceinline__ void gemm_dist(SmemT& sm, int wv, int lane) {
  const int half = lane >> 4, l16 = lane & 15;
  v8f acc0 = {}; v8f acc1 = {};
  const int kb0 = wv * 64;
#pragma unroll
  for (int kk = 0; kk < 64; kk += 4) {
    int k = kb0 + kk + 2 * half;
    v2f bx = *(const v2f*)&sm.xs[l16 * SP + k];              // B: x^T fragment
    v2f a0 = *(const v2f*)&sm.smp[l16 * SP + k];             // A rows 0..15
    v2f a1 = *(const v2f*)&sm.smp[(16 + l16) * SP + k];      // A rows 16..31
    acc0 = wmma4(a0, bx, acc0);
    acc1 = wmma4(a1, bx, acc1);
  }
  if (wv == 0) { // cond GEMM, K = 16 only
    v8f c0 = {}; v8f c1 = {};
#pragma unroll
    for (int kk = 0; kk < DC_N; kk += 4) {
      int k = kk + 2 * half;
      v2f bx = *(const v2f*)&sm.condx[l16 * CP + k];
      v2f a0 = *(const v2f*)&sm.cd[l16 * CP + k];
      v2f a1 = *(const v2f*)&sm.cd[(16 + l16) * CP + k];
      c0 = wmma4(a0, bx, c0);
      c1 = wmma4(a1, bx, c1);
    }
#pragma unroll
    for (int r = 0; r < 8; ++r) {
      sm.Sc[(r + 8 * half) * BT + l16]        = c0[r];
      sm.Sc[(16 + r + 8 * half) * BT + l16]   = c1[r];
    }
  }
#pragma unroll
  for (int r = 0; r < 8; ++r) { // combine per-wave K partials
    unsafeAtomicAdd(&sm.S[(r + 8 * half) * BT + l16],      acc0[r]);
    unsafeAtomicAdd(&sm.S[(16 + r + 8 * half) * BT + l16], acc1[r]);
  }
}

// ---- kernels -----------------------------------------------------------

__global__ void k0_zero(float* __restrict__ p, int n4) {
  int i = blockIdx.x * 256 + threadIdx.x;
  if (i < n4) ((float4*)p)[i] = make_float4(0.f, 0.f, 0.f, 0.f);
}

__global__ __launch_bounds__(256) void k1_stats(
    const float* __restrict__ x, const float* __restrict__ sigma,
    const float* __restrict__ cond, const float* __restrict__ samples,
    const float* __restrict__ cond_data,
    float* __restrict__ pmax, float* __restrict__ psum) {
  __shared__ SmemT sm;
  const int t = threadIdx.x;
  const int bt = blockIdx.x, ch = blockIdx.y;
  const int b0 = bt * BT;
#pragma unroll
  for (int i = 0; i < 8; ++i) { // x tile 16x512
    int fi = i * 256 + t, row = fi >> 7, c4 = fi & 127;
    *(float4*)&sm.xs[row * SP + c4 * 4] =
        *(const float4*)(x + (size_t)(b0 + row) * D_N + c4 * 4);
  }
  if (t < 64) {
    int row = t >> 2, c4 = t & 3;
    *(float4*)&sm.condx[row * CP + c4 * 4] =
        *(const float4*)(cond + (size_t)(b0 + row) * DC_N + c4 * 4);
  }
  if (t < BT) { float s = sigma[b0 + t]; sm.inv2s2[t] = 1.0f / (2.0f * s * s); }

  float mx = -3.0e38f, sum = 0.0f;
  const int mt0 = ch * CPT1, mt1 = min(NMT, mt0 + CPT1);
  const int wv = t >> 5, lane = t & 31;
  for (int mt = mt0; mt < mt1; ++mt) {
    const int m0 = mt * MT;
    if (mt + 1 < mt1) { // prefetch next tile into caches (global_prefetch_b8)
#pragma unroll
      for (int i = 0; i < 4; ++i)
        __builtin_prefetch(samples + (size_t)(m0 + MT) * D_N + (i * 256 + t) * 16, 0, 0);
    }
    __syncthreads();                       // prior consumers done
    load_tiles(sm, samples, cond_data, m0, t);
    __syncthreads();                       // tile + zeros visible
    compute_norms(sm, t);
    gemm_dist(sm, wv, lane);
    __syncthreads();                       // S, Sc, norms complete
    if (t < BT) {                          // online softmax stats, column t
      const float iv = sm.inv2s2[t];
      const int nv = min(MT, M_N - m0);
      for (int j = 0; j < nv; ++j) {
        float lw = (2.0f * sm.S[j * BT + t] - sm.a2[j]) * iv
                 + (2.0f * sm.Sc[j * BT + t] - sm.ca2[j]) * 50.0f; // 1/(2*0.1^2)
        float nm = fmaxf(mx, lw);
        sum = sum * __expf(mx - nm) + __expf(lw - nm);
        mx = nm;
      }
    }
  }
  if (t < BT) {
    pmax[ch * B_N + b0 + t] = mx;
    psum[ch * B_N + b0 + t] = sum;
  }
}

__global__ void k2_reduce(const float* __restrict__ pmax, const float* __restrict__ psum,
                          float* __restrict__ fmax, float* __restrict__ finvZ) {
  int b = blockIdx.x * 256 + threadIdx.x;
  if (b >= B_N) return;
  float mx = -3.0e38f;
  for (int c = 0; c < NCH1; ++c) mx = fmaxf(mx, pmax[c * B_N + b]);
  float z = 0.0f;
  for (int c = 0; c < NCH1; ++c) z += psum[c * B_N + b] * __expf(pmax[c * B_N + b] - mx);
  fmax[b] = mx;
  finvZ[b] = 1.0f / z;
}

__global__ __launch_bounds__(256) void k3_accum_sig(
    const float* __restrict__ x, const float* __restrict__ sigma,
    const float* __restrict__ cond, const float* __restrict__ samples,
    const float* __restrict__ cond_data, const float* __restrict__ fmax,
    const float* __restrict__ finvZ, float* __restrict__ x0out) {
  __shared__ SmemT sm;
  const int t = threadIdx.x;
  const int bt = blockIdx.x, ch = blockIdx.y;
  const int b0 = bt * BT;
#pragma unroll
  for (int i = 0; i < 8; ++i) {
    int fi = i * 256 + t, row = fi >> 7, c4 = fi & 127;
    *(float4*)&sm.xs[row * SP + c4 * 4] =
        *(const float4*)(x + (size_t)(b0 + row) * D_N + c4 * 4);
  }
  if (t < 64) {
    int row = t >> 2, c4 = t & 3;
    *(float4*)&sm.condx[row * CP + c4 * 4] =
        *(const float4*)(cond + (size_t)(b0 + row) * DC_N + c4 * 4);
  }
  if (t < BT) {
    float s = sigma[b0 + t];
    sm.inv2s2[t] = 1.0f / (2.0f * s * s);
    sm.fmaxs[t] = fmax[b0 + t];
    sm.finvZ[t] = finvZ[b0 + t];
  }
  const int wv = t >> 5, lane = t & 31, half = lane >> 4, l16 = lane & 15;
  v8f c2[4] = { v8f{}, v8f{}, v8f{}, v8f{} };  // x0 accum: 4 tiles of [16b x 16d]
  const int mt0 = ch * CPT3, mt1 = min(NMT, mt0 + CPT3);
  for (int mt = mt0; mt < mt1; ++mt) {
    const int m0 = mt * MT;
    if (mt + 1 < mt1) {
#pragma unroll
      for (int i = 0; i < 4; ++i)
        __builtin_prefetch(samples + (size_t)(m0 + MT) * D_N + (i * 256 + t) * 16, 0, 0);
    }
    __syncthreads();
    load_tiles(sm, samples, cond_data, m0, t);
    __syncthreads();
    compute_norms(sm, t);
    gemm_dist(sm, wv, lane);
    __syncthreads();
    // weights -> Wt[b][j]
    for (int e = t; e < MT * BT; e += 256) {
      int j = e >> 4, b = e & 15;
      float w = 0.0f;
      if (m0 + j < M_N) {
        float lw = (2.0f * sm.S[e] - sm.a2[j]) * sm.inv2s2[b]
                 + (2.0f * sm.Sc[e] - sm.ca2[j]) * 50.0f;
        w = __expf(lw - sm.fmaxs[b]) * sm.finvZ[b];
      }
      sm.Wt[b * WJP + j] = w;
    }
    __syncthreads();
    // x0 GEMM: acc[b][d] += sum_j Wt[b][j] * smp[j][d]; wave owns d-slice [64*wv, 64*wv+64)
    const int dbase = wv * 64;
#pragma unroll
    for (int kb = 0; kb < MT; kb += 4) {
      int k = kb + 2 * half;
      v2f wa = *(const v2f*)&sm.Wt[l16 * WJP + k];
#pragma unroll
      for (int c = 0; c < 4; ++c) {
        int dc = dbase + c * 16 + l16;
        v2f bb;
        bb.x = sm.smp[k * SP + dc];
        bb.y = sm.smp[(k + 1) * SP + dc];
        c2[c] = wmma4(wa, bb, c2[c]);
      }
    }
  }
  // accumulate partial x0 into d_out (zeroed by k0) via global_atomic_add_f32
  const int dbase = wv * 64;
#pragma unroll
  for (int c = 0; c < 4; ++c) {
    int d = dbase + c * 16 + l16;
#pragma unroll
    for (int r = 0; r < 8; ++r) {
      int b = r + 8 * half;
      unsafeAtomicAdd(&x0out[(size_t)(b0 + b) * D_N + d], c2[c][r]);
    }
  }
}

__global__ void k4_final(const float* __restrict__ x, const float* __restrict__ sigma,
                         float* __restrict__ out) {
  int i4 = blockIdx.x * 256 + threadIdx.x;
  if (i4 >= B_N * D_N / 4) return;
  int b = i4 >> 7;  // 128 float4 per row of D=512
  float inv = 1.0f / sigma[b];
  float4 xv = ((const float4*)x)[i4];
  float4 x0 = ((float4*)out)[i4];
  float4 r;
  r.x = (xv.x - x0.x) * inv;
  r.y = (xv.y - x0.y) * inv;
  r.z = (xv.z - x0.z) * inv;
  r.w = (xv.w - x0.w) * inv;
  ((float4*)out)[i4] = r;
}

extern "C" void kernel_launch(void* const* d_in, const int* in_sizes, int n_in,
                              void* d_out, int out_size, void* d_ws, size_t ws_size,
                              hipStream_t stream) {
  (void)in_sizes; (void)n_in; (void)out_size; (void)ws_size;
  const float* x         = (const float*)d_in[0];
  const float* sigma     = (const float*)d_in[1];
  const float* cond      = (const float*)d_in[2];
  const float* samples   = (const float*)d_in[3];
  const float* cond_data = (const float*)d_in[4];
  float* out = (float*)d_out;

  float* ws    = (float*)d_ws;
  float* pmax  = ws;                    // [NCH1 * B]
  float* psum  = pmax + NCH1 * B_N;     // [NCH1 * B]
  float* fmax  = psum + NCH1 * B_N;     // [B]
  float* finvZ = fmax + B_N;            // [B]

  const int n4 = B_N * D_N / 4;
  k0_zero<<<(n4 + 255) / 256, 256, 0, stream>>>(out, n4);
  dim3 g1(B_N / BT, NCH1);
  k1_stats<<<g1, 256, 0, stream>>>(x, sigma, cond, samples, cond_data, pmax, psum);
  k2_reduce<<<(B_N + 255) / 256, 256, 0, stream>>>(pmax, psum, fmax, finvZ);
  dim3 g3(B_N / BT, NCH3);
  k3_accum_sig<<<g3, 256, 0, stream>>>(x, sigma, cond, samples, cond_data, fmax, finvZ, out);
  k4_final<<<(n4 + 255) / 256, 256, 0, stream>>>(x, sigma, out);
}